// EdgeUpdaterFrameDiff_20985210208287
// MI455X (gfx1250) — compile-verified
//
#include <hip/hip_runtime.h>

typedef __attribute__((ext_vector_type(2))) float v2f;
typedef __attribute__((ext_vector_type(8))) float v8f;

#define N_NODES  768
#define NODE_DIM 128
#define EDGE_DIM 64
#define HALF     64
#define MLP_DIM  192
#define ZSTRIDE  196   // LDS row stride in floats (192 padded; 196 % 64 != 0 -> conflict-free)
#define OSTRIDE  68    // LDS row stride for 64-wide output tile
#define MTILES   4     // M-tiles (of 16 rows) per block -> 64 rows/block
#define ROWS     (16 * MTILES)
#define LN_EPS   1e-5f

// ---------------------------------------------------------------------------
// Stage 1: x_down[i][o] = sum_k x[i][k] * Wd[k][o] + bd[o]   (768 x 64, tiny)
// ---------------------------------------------------------------------------
__global__ __launch_bounds__(HALF) void xdown_kernel(
    const float* __restrict__ x, const float* __restrict__ Wd,
    const float* __restrict__ bd, float* __restrict__ xd) {
  const int i = blockIdx.x;   // node index
  const int o = threadIdx.x;  // output dim 0..63
  const float* xr = x + (size_t)i * NODE_DIM;
  float acc = bd[o];
#pragma unroll 8
  for (int k = 0; k < NODE_DIM; ++k)
    acc = fmaf(xr[k], Wd[(size_t)k * HALF + o], acc);
  xd[(size_t)i * HALF + o] = acc;
}

// ---------------------------------------------------------------------------
// Stage 2: fused edge MLP.
// One block = 64 consecutive edge rows m = i*N + j  (i constant: 768 % 64 == 0).
// 384 threads = 12 waves; wave w owns output columns [16w, 16w+16) and computes
// MTILES=4 16x16 output tiles, reusing each B fragment across all 4 A tiles
// (4x less L2 weight traffic per WMMA; inner loop 6 loads : 4 wmma).
// All three GEMMs use V_WMMA_F32_16X16X4_F32 (exact fp32 path on CDNA5).
//
// WMMA f32 16x16x4 per-lane fragment layout (CDNA5 ISA 7.12.2):
//   A (16x4):  lanes 0-15: M=lane,   v.x=K0, v.y=K1
//              lanes16-31: M=lane-16,v.x=K2, v.y=K3
//   B (4x16):  lanes 0-15: N=lane,   v.x=K0 row, v.y=K1 row
//              lanes16-31: N=lane-16,v.x=K2 row, v.y=K3 row
//   C/D:       VGPR d: lanes 0-15 -> (M=d, N=lane); lanes16-31 -> (M=d+8, N=lane-16)
// ---------------------------------------------------------------------------
__global__ __launch_bounds__(384) void edge_kernel(
    const float* __restrict__ xd, const float* __restrict__ z,
    const float* __restrict__ W1, const float* __restrict__ b1,
    const float* __restrict__ W2, const float* __restrict__ b2,
    const float* __restrict__ Wo, const float* __restrict__ bo,
    const float* __restrict__ ln_g, const float* __restrict__ ln_b,
    float* __restrict__ out) {
  __shared__ float zin[ROWS * ZSTRIDE];  // z_in tile, later residual s = relu(h2)+z_in
  __shared__ float h1s[ROWS * ZSTRIDE];  // hidden tile
  __shared__ float ots[ROWS * OSTRIDE];  // pre-LN output tile
  __shared__ float mus[ROWS];
  __shared__ float rss[ROWS];

  const int tid  = threadIdx.x;
  const int wave = tid >> 5;
  const int lane = tid & 31;
  const int hi   = lane >> 4;   // 0: lanes 0-15, 1: lanes 16-31
  const int lm   = lane & 15;   // A-frag row / D-frag column sublane

  const int r0     = blockIdx.x * ROWS;  // first flat edge-row of this block
  const int i_node = r0 / N_NODES;       // constant over block (768 % 64 == 0)
  const int j0     = r0 % N_NODES;

  // ---- build z_in tile [64][192] in LDS (coalesced: j contiguous) ----
  for (int idx = tid; idx < ROWS * MLP_DIM; idx += 384) {
    const int lr = idx / MLP_DIM;
    const int c  = idx % MLP_DIM;
    float v;
    if (c < HALF)
      v = xd[(size_t)i_node * HALF + c];                     // row broadcast
    else if (c < 2 * HALF)
      v = xd[(size_t)(j0 + lr) * HALF + (c - HALF)];         // col broadcast
    else
      v = z[((size_t)(i_node * N_NODES + j0 + lr)) * EDGE_DIM + (c - 2 * HALF)];
    zin[lr * ZSTRIDE + c] = v;
  }
  __syncthreads();

  // ---- GEMM1: h1 = relu(z_in @ W1 + b1)  (bias folded into acc init) ----
  {
    const int n = wave * 16 + lm;       // my output column in [0,192)
    const float bv = b1[n];
    v8f acc[MTILES];
#pragma unroll
    for (int t = 0; t < MTILES; ++t) acc[t] = (v8f){bv, bv, bv, bv, bv, bv, bv, bv};
#pragma unroll 2
    for (int kt = 0; kt < MLP_DIM / 4; ++kt) {
      const int k = kt * 4 + 2 * hi;
      v2f b;
      b.x = W1[(size_t)k * MLP_DIM + n];
      b.y = W1[(size_t)(k + 1) * MLP_DIM + n];
#pragma unroll
      for (int t = 0; t < MTILES; ++t) {
        v2f a;
        a.x = zin[(t * 16 + lm) * ZSTRIDE + k];
        a.y = zin[(t * 16 + lm) * ZSTRIDE + k + 1];
        acc[t] = __builtin_amdgcn_wmma_f32_16x16x4_f32(
            false, a, false, b, (short)0, acc[t], false, false);
      }
    }
#pragma unroll
    for (int t = 0; t < MTILES; ++t)
#pragma unroll
      for (int d = 0; d < 8; ++d)
        h1s[(t * 16 + d + 8 * hi) * ZSTRIDE + n] = fmaxf(acc[t][d], 0.0f);
  }
  __syncthreads();

  // ---- GEMM2: s = relu(h1 @ W2 + b2) + z_in, written back into zin ----
  // (columns n are exclusive to this wave -> race-free read-modify-write)
  {
    const int n = wave * 16 + lm;
    const float bv = b2[n];
    v8f acc[MTILES];
#pragma unroll
    for (int t = 0; t < MTILES; ++t) acc[t] = (v8f){bv, bv, bv, bv, bv, bv, bv, bv};
#pragma unroll 2
    for (int kt = 0; kt < MLP_DIM / 4; ++kt) {
      const int k = kt * 4 + 2 * hi;
      v2f b;
      b.x = W2[(size_t)k * MLP_DIM + n];
      b.y = W2[(size_t)(k + 1) * MLP_DIM + n];
#pragma unroll
      for (int t = 0; t < MTILES; ++t) {
        v2f a;
        a.x = h1s[(t * 16 + lm) * ZSTRIDE + k];
        a.y = h1s[(t * 16 + lm) * ZSTRIDE + k + 1];
        acc[t] = __builtin_amdgcn_wmma_f32_16x16x4_f32(
            false, a, false, b, (short)0, acc[t], false, false);
      }
    }
#pragma unroll
    for (int t = 0; t < MTILES; ++t)
#pragma unroll
      for (int d = 0; d < 8; ++d) {
        const int rr = t * 16 + d + 8 * hi;
        zin[rr * ZSTRIDE + n] = fmaxf(acc[t][d], 0.0f) + zin[rr * ZSTRIDE + n];
      }
  }
  __syncthreads();

  // ---- GEMM3: out = s @ Wo + bo  (64 cols -> waves 0..3; branch is wave-uniform) ----
  if (wave < 4) {
    const int n = wave * 16 + lm;       // column in [0,64)
    const float bv = bo[n];
    v8f acc[MTILES];
#pragma unroll
    for (int t = 0; t < MTILES; ++t) acc[t] = (v8f){bv, bv, bv, bv, bv, bv, bv, bv};
#pragma unroll 2
    for (int kt = 0; kt < MLP_DIM / 4; ++kt) {
      const int k = kt * 4 + 2 * hi;
      v2f b;
      b.x = Wo[(size_t)k * EDGE_DIM + n];
      b.y = Wo[(size_t)(k + 1) * EDGE_DIM + n];
#pragma unroll
      for (int t = 0; t < MTILES; ++t) {
        v2f a;
        a.x = zin[(t * 16 + lm) * ZSTRIDE + k];
        a.y = zin[(t * 16 + lm) * ZSTRIDE + k + 1];
        acc[t] = __builtin_amdgcn_wmma_f32_16x16x4_f32(
            false, a, false, b, (short)0, acc[t], false, false);
      }
    }
#pragma unroll
    for (int t = 0; t < MTILES; ++t)
#pragma unroll
      for (int d = 0; d < 8; ++d)
        ots[(t * 16 + d + 8 * hi) * OSTRIDE + n] = acc[t][d];
  }
  __syncthreads();

  // ---- LayerNorm over the 64 output features (two-pass, one row per thread) ----
  if (tid < ROWS) {
    float s = 0.0f;
#pragma unroll 8
    for (int c = 0; c < EDGE_DIM; ++c) s += ots[tid * OSTRIDE + c];
    const float m = s * (1.0f / EDGE_DIM);
    float v = 0.0f;
#pragma unroll 8
    for (int c = 0; c < EDGE_DIM; ++c) {
      const float d = ots[tid * OSTRIDE + c] - m;
      v = fmaf(d, d, v);
    }
    mus[tid] = m;
    rss[tid] = rsqrtf(v * (1.0f / EDGE_DIM) + LN_EPS);
  }
  __syncthreads();

  for (int idx = tid; idx < ROWS * EDGE_DIM; idx += 384) {
    const int lr = idx >> 6;
    const int c  = idx & 63;
    const float val =
        (ots[lr * OSTRIDE + c] - mus[lr]) * rss[lr] * ln_g[c] + ln_b[c];
    out[((size_t)(r0 + lr)) * EDGE_DIM + c] = val;
  }
}

// ---------------------------------------------------------------------------
extern "C" void kernel_launch(void* const* d_in, const int* in_sizes, int n_in,
                              void* d_out, int out_size, void* d_ws, size_t ws_size,
                              hipStream_t stream) {
  (void)in_sizes; (void)n_in; (void)out_size; (void)ws_size;
  const float* x    = (const float*)d_in[0];
  const float* z    = (const float*)d_in[1];
  const float* Wd   = (const float*)d_in[2];
  const float* bd   = (const float*)d_in[3];
  const float* W1   = (const float*)d_in[4];
  const float* b1   = (const float*)d_in[5];
  const float* W2   = (const float*)d_in[6];
  const float* b2   = (const float*)d_in[7];
  const float* Wo   = (const float*)d_in[8];
  const float* bo   = (const float*)d_in[9];
  const float* ln_g = (const float*)d_in[10];
  const float* ln_b = (const float*)d_in[11];
  float* out = (float*)d_out;
  float* xdn = (float*)d_ws;  // 768*64 f32 = 192 KiB scratch

  xdown_kernel<<<N_NODES, HALF, 0, stream>>>(x, Wd, bd, xdn);

  const int blocks = (N_NODES * N_NODES) / ROWS;  // 9216 blocks of 64 edge rows
  edge_kernel<<<blocks, 384, 0, stream>>>(xdn, z, W1, b1, W2, b2, Wo, bo,
                                          ln_g, ln_b, out);
}